// DistanceContainedConv3d_43568148251274
// MI455X (gfx1250) — compile-verified
//
#include <hip/hip_runtime.h>
#include <math.h>

// DistanceContainedConv3d for MI455X (gfx1250, wave32).
// Pipeline: knn -> coeff pre-swizzle -> geometry/eigh/basis -> fused (M1 + WMMA GEMM).
// Dominant GEMM (10000 x 64 x 1728, fp32) runs on V_WMMA_F32_16X16X4_F32.

#define K_NB 16
#define NQ   27          // DEG_N*DEG_L*DEG_M
#define CTOT 1728        // 64 * 27
#define KT_STEPS 432     // 1728 / 4  (K=4 per f32 WMMA)

typedef __attribute__((ext_vector_type(2))) float v2f;
typedef __attribute__((ext_vector_type(8))) float v8f;

// ---------------------------------------------------------------------------
// Kernel 1: brute-force KNN (top-16 smallest squared distance, self included).
// Positions streamed through LDS in chunks; per-thread register top-k with
// max-slot replacement (order irrelevant: downstream reductions over k are
// permutation invariant).
// ---------------------------------------------------------------------------
#define KNN_CHUNK 2048
__global__ void knn_kernel(const float* __restrict__ pos, int* __restrict__ nbr, int S) {
    __shared__ float lpos[KNN_CHUNK * 3];
    const int t = threadIdx.x;
    const int s = blockIdx.x * blockDim.x + t;
    const bool active = (s < S);
    const int sq = active ? s : 0;
    const float qx = pos[sq * 3 + 0], qy = pos[sq * 3 + 1], qz = pos[sq * 3 + 2];

    float bd[K_NB]; int bi[K_NB];
#pragma unroll
    for (int u = 0; u < K_NB; ++u) { bd[u] = 3.4e38f; bi[u] = 0; }
    float worst = 3.4e38f; int wslot = 0;

    for (int base = 0; base < S; base += KNN_CHUNK) {
        const int cnt = min(KNN_CHUNK, S - base);
        __syncthreads();
        for (int idx = t; idx < cnt * 3; idx += blockDim.x)
            lpos[idx] = pos[base * 3 + idx];
        __syncthreads();
        for (int jj = 0; jj < cnt; ++jj) {
            const float dx = qx - lpos[jj * 3 + 0];
            const float dy = qy - lpos[jj * 3 + 1];
            const float dz = qz - lpos[jj * 3 + 2];
            const float d2 = dx * dx + dy * dy + dz * dz;
            if (d2 < worst) {
                const int jg = base + jj;
#pragma unroll
                for (int u = 0; u < K_NB; ++u) if (u == wslot) { bd[u] = d2; bi[u] = jg; }
                worst = -1.0f;
#pragma unroll
                for (int u = 0; u < K_NB; ++u) if (bd[u] > worst) { worst = bd[u]; wslot = u; }
            }
        }
    }
    if (active) {
#pragma unroll
        for (int u = 0; u < K_NB; ++u) nbr[s * K_NB + u] = bi[u];
    }
}

// ---------------------------------------------------------------------------
// Kernel 2: pre-swizzle coeff (O,I,27) into per-lane WMMA B-fragments.
// Layout: ctw[((ot*432 + kt)*64) + lane*2 + j] = CtT[4*kt + kk][16*ot + n]
// with n = lane%16, kk = 2*(lane/16)+j  (f32 16x16x4 fragment convention).
// Each wave's k-stream becomes 432 contiguous 256B blocks -> coalesced b64 loads.
// ---------------------------------------------------------------------------
__global__ void prep_coeff_kernel(const float* __restrict__ coeff, float* __restrict__ ctw) {
    const int u = blockIdx.x * blockDim.x + threadIdx.x;
    if (u >= 4 * KT_STEPS * 64) return;
    const int ot   = u / (KT_STEPS * 64);
    const int rem  = u % (KT_STEPS * 64);
    const int kt   = rem >> 6;
    const int l2   = rem & 63;
    const int lane = l2 >> 1;
    const int j    = l2 & 1;
    const int n    = lane & 15;
    const int kk   = ((lane >> 4) << 1) + j;
    const int c    = kt * 4 + kk;       // c = i*27 + q
    const int o    = ot * 16 + n;
    const int i    = c / NQ;
    const int q    = c % NQ;
    ctw[u] = coeff[(o * 64 + i) * NQ + q];
}

// ---------------------------------------------------------------------------
// Kernel 3: geometry. One thread per point: neighbor mean (-> centers output),
// covariance, closed-form 3x3 symmetric eigh, PCA projection, spherical
// coords, 27-term polynomial basis -> ws.
// ---------------------------------------------------------------------------
__device__ inline void eigvec3(float a00, float a01, float a02,
                               float a11, float a12, float a22, float lam,
                               float& vx, float& vy, float& vz) {
    const float r0x = a00 - lam, r0y = a01, r0z = a02;
    const float r1x = a01, r1y = a11 - lam, r1z = a12;
    const float r2x = a02, r2y = a12, r2z = a22 - lam;
    float c0x = r0y * r1z - r0z * r1y, c0y = r0z * r1x - r0x * r1z, c0z = r0x * r1y - r0y * r1x;
    float c1x = r0y * r2z - r0z * r2y, c1y = r0z * r2x - r0x * r2z, c1z = r0x * r2y - r0y * r2x;
    float c2x = r1y * r2z - r1z * r2y, c2y = r1z * r2x - r1x * r2z, c2z = r1x * r2y - r1y * r2x;
    const float n0 = c0x * c0x + c0y * c0y + c0z * c0z;
    const float n1 = c1x * c1x + c1y * c1y + c1z * c1z;
    const float n2 = c2x * c2x + c2y * c2y + c2z * c2z;
    float bx = c0x, by = c0y, bz = c0z, bn = n0;
    if (n1 > bn) { bx = c1x; by = c1y; bz = c1z; bn = n1; }
    if (n2 > bn) { bx = c2x; by = c2y; bz = c2z; bn = n2; }
    const float inv = rsqrtf(bn + 1e-30f);
    vx = bx * inv; vy = by * inv; vz = bz * inv;
}

__global__ void geom_kernel(const float* __restrict__ pos, const int* __restrict__ nbr,
                            float* __restrict__ centers, float* __restrict__ basis, int S) {
    const int s = blockIdx.x * blockDim.x + threadIdx.x;
    if (s >= S) return;
    float lx[K_NB], ly[K_NB], lz[K_NB];
    float cx = 0.f, cy = 0.f, cz = 0.f;
#pragma unroll
    for (int k = 0; k < K_NB; ++k) {
        const int j = nbr[s * K_NB + k];
        lx[k] = pos[j * 3 + 0]; ly[k] = pos[j * 3 + 1]; lz[k] = pos[j * 3 + 2];
        cx += lx[k]; cy += ly[k]; cz += lz[k];
    }
    const float invK = 1.0f / (float)K_NB;
    cx *= invK; cy *= invK; cz *= invK;
    centers[s * 3 + 0] = cx; centers[s * 3 + 1] = cy; centers[s * 3 + 2] = cz;

    float a00 = 0, a01 = 0, a02 = 0, a11 = 0, a12 = 0, a22 = 0;
#pragma unroll
    for (int k = 0; k < K_NB; ++k) {
        lx[k] -= cx; ly[k] -= cy; lz[k] -= cz;
        a00 += lx[k] * lx[k]; a01 += lx[k] * ly[k]; a02 += lx[k] * lz[k];
        a11 += ly[k] * ly[k]; a12 += ly[k] * lz[k]; a22 += lz[k] * lz[k];
    }
    a00 *= invK; a01 *= invK; a02 *= invK; a11 *= invK; a12 *= invK; a22 *= invK;

    // Closed-form symmetric 3x3 eigenvalues (ascending, matching eigh).
    const float p1 = a01 * a01 + a02 * a02 + a12 * a12;
    const float qm = (a00 + a11 + a22) * (1.0f / 3.0f);
    const float b00 = a00 - qm, b11 = a11 - qm, b22 = a22 - qm;
    const float p2 = b00 * b00 + b11 * b11 + b22 * b22 + 2.0f * p1;
    float v0x, v0y, v0z, v1x, v1y, v1z, v2x, v2y, v2z;
    if (p2 < 1e-14f) {                      // isotropic: any orthonormal basis
        v0x = 1; v0y = 0; v0z = 0; v1x = 0; v1y = 1; v1z = 0; v2x = 0; v2y = 0; v2z = 1;
    } else {
        const float p = sqrtf(p2 * (1.0f / 6.0f));
        const float ip = 1.0f / p;
        const float c00 = b00 * ip, c01 = a01 * ip, c02 = a02 * ip;
        const float c11 = b11 * ip, c12 = a12 * ip, c22 = b22 * ip;
        float r = 0.5f * (c00 * (c11 * c22 - c12 * c12)
                        - c01 * (c01 * c22 - c12 * c02)
                        + c02 * (c01 * c12 - c11 * c02));
        r = fminf(1.0f, fmaxf(-1.0f, r));
        const float phi = acosf(r) * (1.0f / 3.0f);
        const float lam2 = qm + 2.0f * p * cosf(phi);                   // largest
        const float lam0 = qm + 2.0f * p * cosf(phi + 2.0943951023932f); // smallest
        eigvec3(a00, a01, a02, a11, a12, a22, lam0, v0x, v0y, v0z);
        eigvec3(a00, a01, a02, a11, a12, a22, lam2, v2x, v2y, v2z);
        v1x = v2y * v0z - v2z * v0y;                                     // middle = cross
        v1y = v2z * v0x - v2x * v0z;
        v1z = v2x * v0y - v2y * v0x;
        const float in1 = rsqrtf(v1x * v1x + v1y * v1y + v1z * v1z + 1e-30f);
        v1x *= in1; v1y *= in1; v1z *= in1;
    }

    float* bs = basis + (size_t)s * K_NB * NQ;
#pragma unroll
    for (int k = 0; k < K_NB; ++k) {
        const float px = lx[k] * v0x + ly[k] * v0y + lz[k] * v0z;
        const float py = lx[k] * v1x + ly[k] * v1y + lz[k] * v1z;
        const float pz = lx[k] * v2x + ly[k] * v2y + lz[k] * v2z;
        const float r  = sqrtf(px * px + py * py + pz * pz + 1e-8f);
        float ct = pz / r;
        ct = fminf(1.0f - 1e-7f, fmaxf(-1.0f + 1e-7f, ct));
        const float theta = acosf(ct);
        const float phia  = atan2f(py, px);
        const float pr[3] = { 1.0f, r, r * r };
        const float pt[3] = { 1.0f, theta, theta * theta };
        const float pp[3] = { 1.0f, phia, phia * phia };
#pragma unroll
        for (int n = 0; n < 3; ++n)
#pragma unroll
            for (int l = 0; l < 3; ++l)
#pragma unroll
                for (int m = 0; m < 3; ++m)
                    bs[k * NQ + n * 9 + l * 3 + m] = pr[n] * pt[l] * pp[m];
    }
}

// ---------------------------------------------------------------------------
// Kernel 4 (fused): per 16-point tile,
//  step A: M1[s, i*27+q] = sum_k feats[s,k,i]*basis[s,k,q], written straight
//          into LDS in WMMA A-fragment layout (lane = m + 16*(kk>>1), j = kk&1)
//  step B: out(16x64) = M1(16x1728) @ CtT(1728x64) via 4 waves x 432
//          v_wmma_f32_16x16x4_f32, fp32 accumulators resident in VGPRs.
// Requires S % 16 == 0 (S = 10000 here).
// ---------------------------------------------------------------------------
__global__ void conv_kernel(const float* __restrict__ chan, const int* __restrict__ nbr,
                            const float* __restrict__ basis, const float* __restrict__ ctw,
                            float* __restrict__ out) {
    extern __shared__ float smem[];
    int*   lnbr   = (int*)smem;                 // 256 ints
    float* lbasis = smem + 256;                 // 16*16*27 = 6912 floats
    float* a16    = lbasis + 16 * K_NB * NQ;    // 432*64  = 27648 floats (A fragments)

    const int t  = threadIdx.x;
    const int s0 = blockIdx.x * 16;

    lnbr[t] = nbr[s0 * K_NB + t];               // 16 points x 16 neighbors
    for (int idx = t; idx < 16 * K_NB * NQ; idx += 256)
        lbasis[idx] = basis[(size_t)s0 * K_NB * NQ + idx];
    __syncthreads();

    const int sl = t >> 4;                       // local point 0..15
    const int j  = t & 15;
    const float* brow = lbasis + sl * K_NB * NQ;
#pragma unroll
    for (int ii = 0; ii < 4; ++ii) {
        const int i = j + 16 * ii;               // in-channel
        float fr[K_NB];
#pragma unroll
        for (int k = 0; k < K_NB; ++k)
            fr[k] = chan[(size_t)lnbr[sl * K_NB + k] * 64 + i];  // coalesced per half-wave
        for (int q = 0; q < NQ; ++q) {
            float acc = 0.f;
#pragma unroll
            for (int k = 0; k < K_NB; ++k)
                acc = fmaf(fr[k], brow[k * NQ + q], acc);
            const int c  = i * NQ + q;           // 0..1727
            const int kt = c >> 2;
            const int r  = c & 3;
            a16[kt * 64 + ((sl + ((r >> 1) << 4)) << 1) + (r & 1)] = acc;
        }
    }
    __syncthreads();

    if (t < 128) {                               // waves 0..3, wave-uniform (EXEC all ones)
        const int w    = t >> 5;                 // o-tile
        const int lane = t & 31;
        v8f acc = {};
        const float* arow  = a16 + lane * 2;
        const float* bbase = ctw + (size_t)(w * KT_STEPS) * 64 + lane * 2;
        for (int kt = 0; kt < KT_STEPS; ++kt) {
            v2f a; a[0] = arow[kt * 64 + 0]; a[1] = arow[kt * 64 + 1];
            const v2f b = *(const v2f*)(bbase + kt * 64);
            acc = __builtin_amdgcn_wmma_f32_16x16x4_f32(
                false, a, false, b, (short)0, acc, false, false);
        }
        const int n  = lane & 15;
        const int mh = (lane >> 4) << 3;
#pragma unroll
        for (int r = 0; r < 8; ++r)
            out[(size_t)(s0 + mh + r) * 64 + w * 16 + n] = acc[r];
    }
}

// ---------------------------------------------------------------------------
extern "C" void kernel_launch(void* const* d_in, const int* in_sizes, int n_in,
                              void* d_out, int out_size, void* d_ws, size_t ws_size,
                              hipStream_t stream) {
    const float* pos   = (const float*)d_in[0];   // (S,3)
    const float* chan  = (const float*)d_in[1];   // (S,64)
    const float* coeff = (const float*)d_in[2];   // (64,64,27)
    const int S = in_sizes[0] / 3;                // 10000

    // workspace layout (256B aligned): nbr | basis | ctw
    char* ws = (char*)d_ws;
    size_t off = 0;
    int* nbr = (int*)(ws + off);
    off += (((size_t)S * K_NB * sizeof(int)) + 255) & ~(size_t)255;
    float* basis = (float*)(ws + off);
    off += (((size_t)S * K_NB * NQ * sizeof(float)) + 255) & ~(size_t)255;
    float* ctw = (float*)(ws + off);              // 4*432*64 floats

    float* centers = (float*)d_out;               // first output: (S,3)
    float* outm    = (float*)d_out + (size_t)S * 3; // second output: (S,64)

    const int nblk = (S + 255) / 256;
    knn_kernel<<<dim3(nblk), dim3(256), 0, stream>>>(pos, nbr, S);
    prep_coeff_kernel<<<dim3(432), dim3(256), 0, stream>>>(coeff, ctw);
    geom_kernel<<<dim3(nblk), dim3(256), 0, stream>>>(pos, nbr, centers, basis, S);
    const size_t shmem = 256 * sizeof(int) + (16 * K_NB * NQ + KT_STEPS * 64) * sizeof(float);
    conv_kernel<<<dim3(S / 16), dim3(256), shmem, stream>>>(chan, nbr, basis, ctw, outm);
}